// GNN_68985764708764
// MI455X (gfx1250) — compile-verified
//
#include <hip/hip_runtime.h>
#include <hip/hip_bf16.h>

// N = 100000 nodes, D = 64 features, E = 1250000 edges (read from in_sizes).

typedef float v2f __attribute__((ext_vector_type(2)));
typedef float v8f __attribute__((ext_vector_type(8)));

// ---------------------------------------------------------------------------
// Kernel 1: h = old_g @ W via fp32 WMMA (V_WMMA_F32_16X16X4_F32), one wave per
// 16x16 output tile, K=64 in 16 steps of 4. Also zeroes agg at the same
// indices (agg lives in d_out first half, which is poisoned before timing).
//   blockDim = 128 (4 waves); block b covers row-tile b, col-tiles 0..3.
// ---------------------------------------------------------------------------
__global__ void gnn_gemm_wmma(const float* __restrict__ old_g,
                              const float* __restrict__ W,
                              float* __restrict__ h,      // d_out + N*64 (scratch)
                              float* __restrict__ agg,    // d_out (zeroed here)
                              int N) {
  const int lane = threadIdx.x & 31;
  const int wave = threadIdx.x >> 5;
  const int waveId = blockIdx.x * 4 + wave;
  const int row_tile = waveId >> 2;   // which block of 16 rows
  const int col_tile = waveId & 3;    // which block of 16 cols (D=64 -> 4 tiles)
  if (row_tile * 16 >= N) return;     // wave-uniform: EXEC stays all-ones

  const int m0 = row_tile * 16;
  const int n0 = col_tile * 16;
  const int lo = lane & 15;           // M (for A) / N (for B,C)
  const int hi = lane >> 4;           // selects K pair {k,k+1} vs {k+2,k+3}

  const int row = m0 + lo;            // A-matrix row for this lane
  const int col = n0 + lo;            // B/C/D column for this lane

  v8f c = {};
  #pragma unroll
  for (int k = 0; k < 64; k += 4) {
    const int kk = k + 2 * hi;
    // A 16x4 fp32: lane holds row `row`, K = {kk, kk+1} -> contiguous float2
    v2f a = *(const v2f*)(old_g + (size_t)row * 64 + kk);
    // B 4x16 fp32: VGPR0 = row K=kk, VGPR1 = row K=kk+1, N = lo
    v2f b;
    b.x = W[kk * 64 + col];
    b.y = W[(kk + 1) * 64 + col];
    // 8 args: (neg_a, A, neg_b, B, c_mod, C, reuse_a, reuse_b)
    c = __builtin_amdgcn_wmma_f32_16x16x4_f32(false, a, false, b,
                                              (short)0, c, false, false);
  }

  // C/D layout: VGPR v -> M = m0 + v + 8*hi, N = n0 + lo
  #pragma unroll
  for (int v = 0; v < 8; ++v) {
    const size_t idx = (size_t)(m0 + v + 8 * hi) * 64 + n0 + lo;
    h[idx]   = c[v];
    agg[idx] = 0.0f;   // initialize accumulator for the edge scatter phase
  }
}

// ---------------------------------------------------------------------------
// Kernel 2: per-edge msg = tanh(h[src] + old_g[dst]*w); scatter-add into agg.
// One wave per edge: lane L handles features {2L, 2L+1} as a float2, so the
// whole 256B row is gathered with one global_load_b64 per lane (fully
// coalesced; h/old_g are 25.6MB each -> L2-resident on the 192MB L2).
// Scatter uses the hardware fp32 atomic (global_atomic_add_f32, no-return ->
// STOREcnt; retired by the implicit wait-idle at s_endpgm) — forced via
// inline asm so no CAS-loop fallback is possible.
// ---------------------------------------------------------------------------
__global__ void gnn_edge(const float* __restrict__ h,
                         const float* __restrict__ old_g,
                         const float* __restrict__ ew,
                         const int* __restrict__ src,
                         const int* __restrict__ dst,
                         float* __restrict__ agg,
                         int E) {
  const int lane = threadIdx.x & 31;
  const int e = blockIdx.x * (blockDim.x >> 5) + (threadIdx.x >> 5);
  if (e >= E) return;                  // wave-uniform guard

  const int s = src[e];
  const int r = dst[e];
  const float w = ew[e];

  const float2 hv = *(const float2*)(h + (size_t)s * 64 + 2 * lane);
  const float2 og = *(const float2*)(old_g + (size_t)r * 64 + 2 * lane);

  const float m0 = tanhf(hv.x + og.x * w);
  const float m1 = tanhf(hv.y + og.y * w);

  float* p = agg + (size_t)r * 64 + 2 * lane;
  asm volatile("global_atomic_add_f32 %0, %1, off\n\t"
               "global_atomic_add_f32 %0, %2, off offset:4"
               :
               : "v"(p), "v"(m0), "v"(m1)
               : "memory");
}

// ---------------------------------------------------------------------------
// Kernel 3: history_new = history_db + agg (float4-vectorized), written over
// the staged h in d_out's second half. nodes_new == agg is already in place.
// ---------------------------------------------------------------------------
__global__ void gnn_finish(const float* __restrict__ agg,
                           const float* __restrict__ hist_db,
                           float* __restrict__ hist_out,
                           int n4) {
  const int i = blockIdx.x * blockDim.x + threadIdx.x;
  if (i >= n4) return;
  const float4 a = ((const float4*)agg)[i];
  const float4 b = ((const float4*)hist_db)[i];
  float4 o;
  o.x = a.x + b.x; o.y = a.y + b.y; o.z = a.z + b.z; o.w = a.w + b.w;
  ((float4*)hist_out)[i] = o;
}

extern "C" void kernel_launch(void* const* d_in, const int* in_sizes, int n_in,
                              void* d_out, int out_size, void* d_ws, size_t ws_size,
                              hipStream_t stream) {
  const float* old_g   = (const float*)d_in[0];  // [N,64]
  const float* W       = (const float*)d_in[1];  // [64,64]
  const float* ew      = (const float*)d_in[2];  // [E,1]
  const float* hist_db = (const float*)d_in[3];  // [N,64]
  const int*   src     = (const int*)d_in[4];    // [E]
  const int*   dst     = (const int*)d_in[5];    // [E]

  const int N = in_sizes[0] / 64;
  const int E = in_sizes[4];

  float* nodes_new = (float*)d_out;                       // agg accumulator
  float* hist_new  = (float*)d_out + (size_t)N * 64;      // stages h, then history

  // 1) h = old_g @ W (fp32 WMMA), and zero agg.
  {
    const int rowTiles = (N + 15) / 16;
    gnn_gemm_wmma<<<rowTiles, 128, 0, stream>>>(old_g, W, hist_new, nodes_new, N);
  }

  // 2) Edge gather + tanh + hardware-atomic scatter-add (1 wave / edge).
  {
    const int wavesPerBlock = 8;       // 256 threads
    const int blocks = (E + wavesPerBlock - 1) / wavesPerBlock;
    gnn_edge<<<blocks, 256, 0, stream>>>(hist_new, old_g, ew, src, dst, nodes_new, E);
  }

  // 3) history_new = history_db + agg (overwrites staged h).
  {
    const int n4 = N * 16;  // N*64/4 float4s
    gnn_finish<<<(n4 + 255) / 256, 256, 0, stream>>>(nodes_new, hist_db, hist_new, n4);
  }
}